// TemporalAttentionModule_4303557230858
// MI455X (gfx1250) — compile-verified
//
#include <hip/hip_runtime.h>

// Problem constants (from reference): N=8, C=64, H=W=48
#define NF   8
#define CIN  64
#define C8s  8
#define C2s  32
#define HWs  2304          // 48*48
#define QT   16            // query tile (one WMMA M tile)
#define NQT  (HWs / QT)    // 144 key/query tiles per frame
#define KSL  (HWs / 4)     // 576-key slice per wave in phase 3

typedef __attribute__((ext_vector_type(2))) float v2f;
typedef __attribute__((ext_vector_type(8))) float v8f;

// CDNA5 fp32 WMMA: D(16x16) = A(16x4) * B(4x16) + C   (wave32, fp32 in/out)
__device__ __forceinline__ v8f wmma_f32_k4(v2f a, v2f b, v8f c) {
  return __builtin_amdgcn_wmma_f32_16x16x4_f32(
      /*neg_a=*/false, a, /*neg_b=*/false, b,
      /*c_mod=*/(short)0, c, /*reuse_a=*/false, /*reuse_b=*/false);
}

// ---------------------------------------------------------------------------
// Async global->LDS copy (CDNA5 GLOBAL_LOAD_ASYNC_TO_LDS_B32, ASYNCcnt).
// Probe result: builtin expects address-space(1) int* (global) operands.
// ---------------------------------------------------------------------------
typedef __attribute__((address_space(1))) int as1_int;
typedef __attribute__((address_space(3))) int as3_int;

__device__ __forceinline__ void async_copy_b32(const float* __restrict__ g,
                                               float* __restrict__ l) {
#if defined(__gfx1250__) && __has_builtin(__builtin_amdgcn_global_load_async_to_lds_b32)
  __builtin_amdgcn_global_load_async_to_lds_b32(
      (as1_int*)(g), (as3_int*)(l), 0, 0);
#else
  *l = *g;
#endif
}

__device__ __forceinline__ void wait_async_then_barrier() {
#if defined(__gfx1250__) && __has_builtin(__builtin_amdgcn_global_load_async_to_lds_b32)
#if __has_builtin(__builtin_amdgcn_s_wait_asynccnt)
  __builtin_amdgcn_s_wait_asynccnt(0);
#else
  asm volatile("s_wait_asynccnt 0x0" ::: "memory");
#endif
#endif
  __syncthreads();
}

// ---------------------------------------------------------------------------
// Fused 1x1-conv Q/K/V: q=conv(x,qw)+qb, k=conv(x,kw)+kb, v=relu(conv(x,vw)+vb)
// x: (NF, CIN, HW) ; q,k: (NF, C8, HW) ; v: (NF, C2, HW)
// grid = (HW/256, NF), block = 256. Weights staged to LDS asynchronously.
// ---------------------------------------------------------------------------
__global__ __launch_bounds__(256) void qkv_kernel(
    const float* __restrict__ x,
    const float* __restrict__ qw, const float* __restrict__ qb,
    const float* __restrict__ kw, const float* __restrict__ kb,
    const float* __restrict__ vw, const float* __restrict__ vb,
    float* __restrict__ q, float* __restrict__ k, float* __restrict__ v) {
  __shared__ float swq[C8s * CIN], swk[C8s * CIN], swv[C2s * CIN];
  __shared__ float sbq[C8s], sbk[C8s], sbv[C2s];
  for (int i = threadIdx.x; i < C8s * CIN; i += 256) {
    async_copy_b32(qw + i, swq + i);
    async_copy_b32(kw + i, swk + i);
  }
  for (int i = threadIdx.x; i < C2s * CIN; i += 256) async_copy_b32(vw + i, swv + i);
  if (threadIdx.x < C8s) {
    async_copy_b32(qb + threadIdx.x, sbq + threadIdx.x);
    async_copy_b32(kb + threadIdx.x, sbk + threadIdx.x);
  }
  if (threadIdx.x < C2s) async_copy_b32(vb + threadIdx.x, sbv + threadIdx.x);
  wait_async_then_barrier();

  const int n   = blockIdx.y;
  const int pos = blockIdx.x * 256 + threadIdx.x;
  const float* xp = x + (size_t)n * CIN * HWs + pos;

  float xv[CIN];
#pragma unroll
  for (int c = 0; c < CIN; ++c) xv[c] = xp[(size_t)c * HWs];

  for (int o = 0; o < C8s; ++o) {
    float aq = sbq[o], ak = sbk[o];
#pragma unroll
    for (int c = 0; c < CIN; ++c) {
      aq = fmaf(xv[c], swq[o * CIN + c], aq);
      ak = fmaf(xv[c], swk[o * CIN + c], ak);
    }
    q[(size_t)n * C8s * HWs + (size_t)o * HWs + pos] = aq;
    k[(size_t)n * C8s * HWs + (size_t)o * HWs + pos] = ak;
  }
  for (int o = 0; o < C2s; ++o) {
    float av = sbv[o];
#pragma unroll
    for (int c = 0; c < CIN; ++c) av = fmaf(xv[c], swv[o * CIN + c], av);
    v[(size_t)n * C2s * HWs + (size_t)o * HWs + pos] = fmaxf(av, 0.f);
  }
}

// ---------------------------------------------------------------------------
// Visual spatial attention, one (frame, 16-query tile) per 256-thread block.
// Phase 1: 8 waves compute S = Q·K^T (K=8 via 2x WMMA k4) into LDS (16x2304 f32).
// Phase 2: block-wide softmax; P kept UNnormalized, row sums saved.
// Phase 3: ALL 8 waves compute O = P·V^T: wave w -> N-tile (w&1), key-slice
//          (w>>1) of 576 keys; 16x16 partials reduced through LDS, softmax
//          normalization folded into the reduction epilogue.
// LDS: 147456 (scores) + 8192 (partials) + eps ≈ 157 KB.
// ---------------------------------------------------------------------------
__global__ __launch_bounds__(256) void vis_attn_kernel(
    const float* __restrict__ q, const float* __restrict__ k,
    const float* __restrict__ v, float* __restrict__ o) {
  __shared__ float sc[QT][HWs];
  __shared__ float part[8][QT][16];
  __shared__ float red[QT][16];
  __shared__ float rowmax[QT];
  __shared__ float rowsum[QT];

  const int n    = blockIdx.y;
  const int m0   = blockIdx.x * QT;
  const int tid  = threadIdx.x;
  const int w    = tid >> 5;
  const int lane = tid & 31;
  const int half = lane >> 4;     // 0: lanes 0-15 (K 0,1) ; 1: lanes 16-31 (K 2,3)
  const int l16  = lane & 15;

  const float* qn = q + (size_t)n * C8s * HWs;
  const float* kn = k + (size_t)n * C8s * HWs;

  // A fragments (query tile, M = l16, K-chunks {0..3} and {4..7}) -- reused.
  v2f a0, a1;
  {
    const int mr = m0 + l16;
    const int c0 = half * 2;
    a0.x = qn[(size_t)(c0 + 0) * HWs + mr];
    a0.y = qn[(size_t)(c0 + 1) * HWs + mr];
    a1.x = qn[(size_t)(4 + c0 + 0) * HWs + mr];
    a1.y = qn[(size_t)(4 + c0 + 1) * HWs + mr];
  }

  // Phase 1: scores
  for (int j = w; j < NQT; j += 8) {          // wave-uniform -> EXEC all ones
    const int k0  = j * QT;
    const int col = k0 + l16;                 // B column = key index
    const int c0  = half * 2;
    v2f b0, b1;
    b0.x = kn[(size_t)(c0 + 0) * HWs + col];
    b0.y = kn[(size_t)(c0 + 1) * HWs + col];
    b1.x = kn[(size_t)(4 + c0 + 0) * HWs + col];
    b1.y = kn[(size_t)(4 + c0 + 1) * HWs + col];
    v8f s = {};
    s = wmma_f32_k4(a0, b0, s);
    s = wmma_f32_k4(a1, b1, s);
    const int rbase = half * 8;               // D rows: r (+8 for upper half)
#pragma unroll
    for (int r = 0; r < 8; ++r) sc[rbase + r][col] = s[r];
  }
  __syncthreads();

  // Phase 2: softmax over 2304 keys per row (16 threads per row)
  {
    const int r  = tid >> 4;
    const int c0 = tid & 15;
    float m = -__builtin_inff();
    for (int c = c0; c < HWs; c += 16) m = fmaxf(m, sc[r][c]);
    red[r][c0] = m;
    __syncthreads();
    if (c0 == 0) {
      float mm = red[r][0];
#pragma unroll
      for (int i = 1; i < 16; ++i) mm = fmaxf(mm, red[r][i]);
      rowmax[r] = mm;
    }
    __syncthreads();
    const float mm = rowmax[r];
    float s = 0.f;
    for (int c = c0; c < HWs; c += 16) {
      float e = __expf(sc[r][c] - mm);
      sc[r][c] = e;                 // unnormalized probability
      s += e;
    }
    red[r][c0] = s;
    __syncthreads();
    if (c0 == 0) {
      float ss = 0.f;
#pragma unroll
      for (int i = 0; i < 16; ++i) ss += red[r][i];
      rowsum[r] = ss;
    }
    __syncthreads();
  }

  // Phase 3: O = P · V^T, all 8 waves (wave-uniform loop bounds, EXEC all ones)
  {
    const int ntile = w & 1;                  // output-channel tile (0 or 1)
    const int slice = w >> 1;                 // key-range slice (0..3)
    const int kbeg  = slice * KSL;
    const int kend  = kbeg + KSL;
    const int ch    = ntile * 16 + l16;       // B column = output channel
    const int koff  = half * 2;
    const float* vch = v + (size_t)n * C2s * HWs + (size_t)ch * HWs;
    v8f acc = {};
    for (int k0 = kbeg; k0 < kend; k0 += 4) {
      if ((k0 & 255) == 0 && k0 + 256 < kend)
        __builtin_prefetch(vch + k0 + 256, 0, 1);   // global_prefetch_b8
      v2f a, b;
      a.x = sc[l16][k0 + koff];
      a.y = sc[l16][k0 + koff + 1];
      b.x = vch[k0 + koff];
      b.y = vch[k0 + koff + 1];
      acc = wmma_f32_k4(a, b, acc);
    }
    const int rbase = half * 8;
#pragma unroll
    for (int r = 0; r < 8; ++r) part[w][rbase + r][l16] = acc[r];
  }
  __syncthreads();

  // Cross-wave reduction + softmax normalization + store (one (M,N) per thread,
  // both channel tiles).
  {
    const int M  = tid >> 4;                  // query row within tile
    const int Nn = tid & 15;                  // channel within tile
    const float inv = 1.f / rowsum[M];
    const float s0 = ((part[0][M][Nn] + part[2][M][Nn]) +
                      (part[4][M][Nn] + part[6][M][Nn])) * inv;
    const float s1 = ((part[1][M][Nn] + part[3][M][Nn]) +
                      (part[5][M][Nn] + part[7][M][Nn])) * inv;
    float* on = o + (size_t)n * C2s * HWs;
    on[(size_t)Nn * HWs + m0 + M]        = s0;
    on[(size_t)(16 + Nn) * HWs + m0 + M] = s1;
  }
}

// ---------------------------------------------------------------------------
// Visual epilogue: out = vc_w·attn + vc_b ; stage0 = concat([out, v]) ;
// also writes d_out channel block 0.
// ---------------------------------------------------------------------------
__global__ __launch_bounds__(256) void post_concat_vis_kernel(
    const float* __restrict__ ao, const float* __restrict__ v,
    const float* __restrict__ cw, const float* __restrict__ cb,
    float* __restrict__ stage, float* __restrict__ dout) {
  __shared__ float sw[C2s * C2s];
  __shared__ float sb[C2s];
  for (int i = threadIdx.x; i < C2s * C2s; i += 256) async_copy_b32(cw + i, sw + i);
  if (threadIdx.x < C2s) async_copy_b32(cb + threadIdx.x, sb + threadIdx.x);
  wait_async_then_barrier();

  const int n   = blockIdx.y;
  const int pos = blockIdx.x * 256 + threadIdx.x;

  float iv[C2s];
#pragma unroll
  for (int c = 0; c < C2s; ++c) iv[c] = ao[(size_t)n * C2s * HWs + (size_t)c * HWs + pos];

  float* st = stage + (size_t)n * CIN * HWs;
  float* dn = dout + (size_t)n * (6 * CIN) * HWs;   // block 0: channels [0,64)
  for (int oc = 0; oc < C2s; ++oc) {
    float a = sb[oc];
#pragma unroll
    for (int c = 0; c < C2s; ++c) a = fmaf(iv[c], sw[oc * C2s + c], a);
    st[(size_t)oc * HWs + pos] = a;
    dn[(size_t)oc * HWs + pos] = a;
  }
  for (int c = 0; c < C2s; ++c) {
    float val = v[(size_t)n * C2s * HWs + (size_t)c * HWs + pos];
    st[(size_t)(C2s + c) * HWs + pos] = val;
    dn[(size_t)(C2s + c) * HWs + pos] = val;
  }
}

// ---------------------------------------------------------------------------
// Temporal head: 3-tap (frames n-2,n-1,n; zero-padded) softmax over logits
// <precede(k), q>, weighted sum of precede(v), 1x1 conv, concat([out, v]).
// Writes stage_next and d_out channel block `blk`.
// ---------------------------------------------------------------------------
__global__ __launch_bounds__(256) void temporal_attn_post_kernel(
    const float* __restrict__ tq, const float* __restrict__ tk,
    const float* __restrict__ tv,
    const float* __restrict__ cw, const float* __restrict__ cb,
    float* __restrict__ stage_next, float* __restrict__ dout, int blk) {
  __shared__ float sw[C2s * C2s];
  __shared__ float sb[C2s];
  for (int i = threadIdx.x; i < C2s * C2s; i += 256) async_copy_b32(cw + i, sw + i);
  if (threadIdx.x < C2s) async_copy_b32(cb + threadIdx.x, sb + threadIdx.x);
  wait_async_then_barrier();

  const int n   = blockIdx.y;
  const int pos = blockIdx.x * 256 + threadIdx.x;

  float qv[C8s];
#pragma unroll
  for (int c = 0; c < C8s; ++c) qv[c] = tq[(size_t)n * C8s * HWs + (size_t)c * HWs + pos];

  float l[3];
#pragma unroll
  for (int t = 0; t < 3; ++t) {
    const int f = n - 2 + t;
    float acc = 0.f;
    if (f >= 0) {
#pragma unroll
      for (int c = 0; c < C8s; ++c)
        acc = fmaf(tk[(size_t)f * C8s * HWs + (size_t)c * HWs + pos], qv[c], acc);
    }
    l[t] = acc;                        // padded frames: k==0 -> logit 0 (matches ref)
  }
  const float m = fmaxf(fmaxf(l[0], l[1]), l[2]);
  float e0 = __expf(l[0] - m), e1 = __expf(l[1] - m), e2 = __expf(l[2] - m);
  const float inv = 1.f / (e0 + e1 + e2);
  const float a[3] = {e0 * inv, e1 * inv, e2 * inv};

  float ov[C2s];
#pragma unroll
  for (int c = 0; c < C2s; ++c) ov[c] = 0.f;
#pragma unroll
  for (int t = 0; t < 3; ++t) {
    const int f = n - 2 + t;
    if (f >= 0) {
#pragma unroll
      for (int c = 0; c < C2s; ++c)
        ov[c] = fmaf(a[t], tv[(size_t)f * C2s * HWs + (size_t)c * HWs + pos], ov[c]);
    }
  }

  float* st = stage_next + (size_t)n * CIN * HWs;
  float* dn = dout + (size_t)n * (6 * CIN) * HWs + (size_t)blk * CIN * HWs;
  for (int oc = 0; oc < C2s; ++oc) {
    float acc = sb[oc];
#pragma unroll
    for (int c = 0; c < C2s; ++c) acc = fmaf(ov[c], sw[oc * C2s + c], acc);
    st[(size_t)oc * HWs + pos] = acc;
    dn[(size_t)oc * HWs + pos] = acc;
  }
  for (int c = 0; c < C2s; ++c) {
    float val = tv[(size_t)n * C2s * HWs + (size_t)c * HWs + pos];
    st[(size_t)(C2s + c) * HWs + pos] = val;
    dn[(size_t)(C2s + c) * HWs + pos] = val;
  }
}

// ---------------------------------------------------------------------------
extern "C" void kernel_launch(void* const* d_in, const int* in_sizes, int n_in,
                              void* d_out, int out_size, void* d_ws, size_t ws_size,
                              hipStream_t stream) {
  (void)in_sizes; (void)n_in; (void)out_size; (void)ws_size;
  const float* x    = (const float*)d_in[0];
  const float* vq_w = (const float*)d_in[1];
  const float* vq_b = (const float*)d_in[2];
  const float* vk_w = (const float*)d_in[3];
  const float* vk_b = (const float*)d_in[4];
  const float* vv_w = (const float*)d_in[5];
  const float* vv_b = (const float*)d_in[6];
  const float* vc_w = (const float*)d_in[7];
  const float* vc_b = (const float*)d_in[8];
  const float* tq_w = (const float*)d_in[9];
  const float* tq_b = (const float*)d_in[10];
  const float* tk_w = (const float*)d_in[11];
  const float* tk_b = (const float*)d_in[12];
  const float* tv_w = (const float*)d_in[13];
  const float* tv_b = (const float*)d_in[14];
  const float* tc_w = (const float*)d_in[15];
  const float* tc_b = (const float*)d_in[16];
  float* out = (float*)d_out;

  // Workspace carve-out (~18.9 MB of fp32)
  float* ws     = (float*)d_ws;
  float* q_vis  = ws; ws += (size_t)NF * C8s * HWs;
  float* k_vis  = ws; ws += (size_t)NF * C8s * HWs;
  float* v_vis  = ws; ws += (size_t)NF * C2s * HWs;
  float* attn_o = ws; ws += (size_t)NF * C2s * HWs;
  float* stageA = ws; ws += (size_t)NF * CIN * HWs;
  float* stageB = ws; ws += (size_t)NF * CIN * HWs;
  float* tq     = ws; ws += (size_t)NF * C8s * HWs;
  float* tk     = ws; ws += (size_t)NF * C8s * HWs;
  float* tv     = ws; ws += (size_t)NF * C2s * HWs;

  const dim3 b256(256);
  const dim3 gpos(HWs / 256, NF);     // (9, 8)
  const dim3 gattn(NQT, NF);          // (144, 8)

  // Visual head
  qkv_kernel<<<gpos, b256, 0, stream>>>(x, vq_w, vq_b, vk_w, vk_b, vv_w, vv_b,
                                        q_vis, k_vis, v_vis);
  vis_attn_kernel<<<gattn, b256, 0, stream>>>(q_vis, k_vis, v_vis, attn_o);
  post_concat_vis_kernel<<<gpos, b256, 0, stream>>>(attn_o, v_vis, vc_w, vc_b,
                                                    stageA, out);
  // Five temporal heads
  float* cur = stageA;
  float* nxt = stageB;
  for (int i = 0; i < 5; ++i) {
    qkv_kernel<<<gpos, b256, 0, stream>>>(cur,
        tq_w + (size_t)i * C8s * CIN, tq_b + (size_t)i * C8s,
        tk_w + (size_t)i * C8s * CIN, tk_b + (size_t)i * C8s,
        tv_w + (size_t)i * C2s * CIN, tv_b + (size_t)i * C2s,
        tq, tk, tv);
    temporal_attn_post_kernel<<<gpos, b256, 0, stream>>>(tq, tk, tv,
        tc_w + (size_t)i * C2s * C2s, tc_b + (size_t)i * C2s,
        nxt, out, i + 1);
    float* t = cur; cur = nxt; nxt = t;
  }
}